// HeteGNN_22050362097670
// MI455X (gfx1250) — compile-verified
//
#include <hip/hip_runtime.h>

typedef __attribute__((ext_vector_type(2))) float v2f;
typedef __attribute__((ext_vector_type(8))) float v8f;

#define D_FEAT 128   // in/hid/out feature width (all 128 here)

// ---------------------------------------------------------------------------
// Utility kernels
// ---------------------------------------------------------------------------
__global__ void zero_kernel(float* __restrict__ p, int n) {
    int i = blockIdx.x * blockDim.x + threadIdx.x;
    if (i < n) p[i] = 0.0f;
}

// Count degrees with float atomics; idx spans R*E, relation = idx / E.
__global__ void degree_kernel(const int* __restrict__ src, const int* __restrict__ dst,
                              float* __restrict__ deg_out, float* __restrict__ deg_in,
                              int total, int E, int N) {
    int i = blockIdx.x * blockDim.x + threadIdx.x;
    if (i >= total) return;
    int r = i / E;
    atomicAdd(&deg_out[(size_t)r * N + src[i]], 1.0f);
    atomicAdd(&deg_in [(size_t)r * N + dst[i]], 1.0f);
}

// deg -> rsqrt(max(deg,1))  (applied to both norm buffers, laid out contiguously)
__global__ void norm_kernel(float* __restrict__ p, int n) {
    int i = blockIdx.x * blockDim.x + threadIdx.x;
    if (i < n) p[i] = rsqrtf(fmaxf(p[i], 1.0f));
}

// out[n][j] = sum_r bias[r][j]   (accumulator init; bias broadcast over rows)
__global__ void init_bias_kernel(float* __restrict__ out, const float* __restrict__ b,
                                 int total, int H, int R) {
    int i = blockIdx.x * blockDim.x + threadIdx.x;
    if (i >= total) return;
    int j = i & (H - 1);
    float s = 0.0f;
    for (int r = 0; r < R; ++r) s += b[r * H + j];
    out[i] = s;
}

__global__ void relu_kernel(float* __restrict__ p, int n) {
    int i = blockIdx.x * blockDim.x + threadIdx.x;
    if (i < n) p[i] = fmaxf(p[i], 0.0f);
}

// ---------------------------------------------------------------------------
// Dense: T[row][col] = norm_s[row] * sum_k X[row][k] * W[k][col]
// One wave computes one 16x16 tile via 32x V_WMMA_F32_16X16X4_F32 (exact fp32).
// 8 waves / block -> block covers 16 rows x 128 cols. N % 16 == 0, so no
// divergence (WMMA requires EXEC all-ones).
// ---------------------------------------------------------------------------
__global__ __launch_bounds__(256) void gemm_norm_wmma(
    const float* __restrict__ X,       // nrows x 128, row major
    const float* __restrict__ W,       // 128 x 128, row major
    const float* __restrict__ norms,   // nrows (norm_s for this relation)
    float* __restrict__ T) {           // nrows x 128
    const int wave  = threadIdx.x >> 5;   // 0..7 -> column tile
    const int lane  = threadIdx.x & 31;
    const int lrow  = lane & 15;          // M index (A) / N index (B,C)
    const int khalf = lane >> 4;          // 0: K{0,1}, 1: K{2,3}
    const int m0 = blockIdx.x * 16;
    const int n0 = wave * 16;

    const float* __restrict__ xrow = X + (size_t)(m0 + lrow) * D_FEAT;
    const int col = n0 + lrow;

    v8f acc = {};
#pragma unroll
    for (int k0 = 0; k0 < D_FEAT; k0 += 4) {
        const int ka = k0 + 2 * khalf;
        v2f a;                                     // A 16x4 fragment
        a.x = xrow[ka];
        a.y = xrow[ka + 1];
        v2f b;                                     // B 4x16 fragment
        b.x = W[(size_t)ka * D_FEAT + col];
        b.y = W[(size_t)(ka + 1) * D_FEAT + col];
        acc = __builtin_amdgcn_wmma_f32_16x16x4_f32(
            /*neg_a=*/false, a, /*neg_b=*/false, b,
            /*c_mod=*/(short)0, acc, /*reuse_a=*/false, /*reuse_b=*/false);
    }

    // C/D layout: VGPR i, lanes 0-15 -> M=i, lanes 16-31 -> M=i+8
    const int rbase = m0 + 8 * khalf;
#pragma unroll
    for (int i = 0; i < 8; ++i) {
        const int row = rbase + i;
        T[(size_t)row * D_FEAT + col] = acc[i] * norms[row];
    }
}

// ---------------------------------------------------------------------------
// Sparse: out[dst[e]][:] += norm_d[dst[e]] * T[src[e]][:]
// One wave per edge: 32 lanes x float4 = the full 512B row, coalesced gather,
// then 4 global f32 atomics per lane (resolved in L2; dst working set = 51MB
// fits the 192MB L2).
// ---------------------------------------------------------------------------
__global__ __launch_bounds__(256) void scatter_kernel(
    const float* __restrict__ T, const int* __restrict__ src,
    const int* __restrict__ dst, const float* __restrict__ norm_d,
    float* __restrict__ out, int E) {
    const int e = blockIdx.x * 8 + (threadIdx.x >> 5);
    if (e >= E) return;
    const int lane = threadIdx.x & 31;
    const int s = src[e];
    const int d = dst[e];
    const float w = norm_d[d];
    const float4 v = reinterpret_cast<const float4*>(T + (size_t)s * D_FEAT)[lane];
    float* o = out + (size_t)d * D_FEAT + lane * 4;
    atomicAdd(o + 0, v.x * w);
    atomicAdd(o + 1, v.y * w);
    atomicAdd(o + 2, v.z * w);
    atomicAdd(o + 3, v.w * w);
}

// ---------------------------------------------------------------------------
// Host-side orchestration
// ---------------------------------------------------------------------------
extern "C" void kernel_launch(void* const* d_in, const int* in_sizes, int n_in,
                              void* d_out, int out_size, void* d_ws, size_t ws_size,
                              hipStream_t stream) {
    const float* x   = (const float*)d_in[0];
    const int*   src = (const int*)  d_in[1];
    const int*   dst = (const int*)  d_in[2];
    const float* W1  = (const float*)d_in[3];
    const float* b1  = (const float*)d_in[4];
    const float* W2  = (const float*)d_in[5];
    const float* b2  = (const float*)d_in[6];
    float* out = (float*)d_out;

    const int H = D_FEAT;
    const int N = in_sizes[0] / H;                 // 100000
    const int R = in_sizes[3] / (H * H);           // 3
    const int E = in_sizes[1] / R;                 // 800000

    // Workspace layout
    float* T        = (float*)d_ws;                // N*128
    float* h1       = T  + (size_t)N * H;          // N*128
    float* norm_out = h1 + (size_t)N * H;          // R*N  (deg_out -> norm_s)
    float* norm_in  = norm_out + (size_t)R * N;    // R*N  (deg_in  -> norm_d)

    const int BLK = 256;
    const int NH  = N * H;

    // --- degrees -> normalization factors (shared by both layers) ---
    zero_kernel<<<dim3((2 * R * N + BLK - 1) / BLK), BLK, 0, stream>>>(norm_out, 2 * R * N);
    degree_kernel<<<dim3((R * E + BLK - 1) / BLK), BLK, 0, stream>>>(
        src, dst, norm_out, norm_in, R * E, E, N);
    norm_kernel<<<dim3((2 * R * N + BLK - 1) / BLK), BLK, 0, stream>>>(norm_out, 2 * R * N);

    // --- layer 1: h1 = sum_r [ norm_d . A_r (norm_s . (x W1_r)) + b1_r ] ---
    init_bias_kernel<<<dim3((NH + BLK - 1) / BLK), BLK, 0, stream>>>(h1, b1, NH, H, R);
    for (int r = 0; r < R; ++r) {
        gemm_norm_wmma<<<dim3(N / 16), BLK, 0, stream>>>(
            x, W1 + (size_t)r * H * H, norm_out + (size_t)r * N, T);
        scatter_kernel<<<dim3((E + 7) / 8), BLK, 0, stream>>>(
            T, src + (size_t)r * E, dst + (size_t)r * E, norm_in + (size_t)r * N, h1, E);
    }
    relu_kernel<<<dim3((NH + BLK - 1) / BLK), BLK, 0, stream>>>(h1, NH);

    // --- layer 2: out = sum_r [ norm_d . A_r (norm_s . (h1 W2_r)) + b2_r ] ---
    init_bias_kernel<<<dim3((NH + BLK - 1) / BLK), BLK, 0, stream>>>(out, b2, NH, H, R);
    for (int r = 0; r < R; ++r) {
        gemm_norm_wmma<<<dim3(N / 16), BLK, 0, stream>>>(
            h1, W2 + (size_t)r * H * H, norm_out + (size_t)r * N, T);
        scatter_kernel<<<dim3((E + 7) / 8), BLK, 0, stream>>>(
            T, src + (size_t)r * E, dst + (size_t)r * E, norm_in + (size_t)r * N, out, E);
    }
}